// RQAutoencoder_5866925326726
// MI455X (gfx1250) — compile-verified
//
#include <hip/hip_runtime.h>
#include <cstdint>

// ---------------------------------------------------------------------------
// RQ autoencoder, MI455X (gfx1250, wave32).
//   enc: Linear(4,2)+ReLU -> Linear(2,1)        (layer0 via V_WMMA_F32_16X16X4_F32)
//   residual VQ: 8 x 1024-entry scalar codebooks (sorted once; binary search in LDS,
//                staged per-block via TDM tensor_load_to_lds)
//   dec: Linear(1,2)+ReLU -> Linear(2,4)
// ---------------------------------------------------------------------------

typedef float v2f  __attribute__((ext_vector_type(2)));
typedef float v8f  __attribute__((ext_vector_type(8)));
typedef unsigned int u32x4 __attribute__((ext_vector_type(4)));
typedef int   i32x8 __attribute__((ext_vector_type(8)));
typedef int   i32x4 __attribute__((ext_vector_type(4)));

#define NQ 8
#define NK 1024
#define CB_ELEMS (NQ * NK)        // 8192 floats = 32 KB

// ------------------------- codebook bitonic sort ---------------------------
// grid = 8 blocks, block = 1024 threads. Sorting is output-equivalent because
// only the nearest *value* is consumed (indices are discarded by reference).
__global__ __launch_bounds__(NK) void rq_sort_codebooks(const float* __restrict__ cb,
                                                        float* __restrict__ sorted) {
  __shared__ float s[NK];
  const int q = blockIdx.x;
  const int i = threadIdx.x;
  s[i] = cb[q * NK + i];
  __syncthreads();
  for (int k = 2; k <= NK; k <<= 1) {
    for (int j = k >> 1; j > 0; j >>= 1) {
      int ixj = i ^ j;
      if (ixj > i) {
        float a = s[i], b = s[ixj];
        bool up = ((i & k) == 0);
        if (up ? (a > b) : (a < b)) { s[i] = b; s[ixj] = a; }
      }
      __syncthreads();
    }
  }
  sorted[q * NK + i] = s[i];
}

// ------------------------------ main kernel --------------------------------
// block = 256 threads (8 waves); each wave owns 32 tokens, one token/lane.
template <bool SORTED>
__global__ __launch_bounds__(256) void rq_autoenc(
    const float* __restrict__ x,
    const float* __restrict__ ew0, const float* __restrict__ eb0,
    const float* __restrict__ ew1, const float* __restrict__ eb1,
    const float* __restrict__ dw0, const float* __restrict__ db0,
    const float* __restrict__ dw1, const float* __restrict__ db1,
    const float* __restrict__ cb,      // sorted (SORTED=1) or raw codebooks
    float* __restrict__ out, int ntok) {
  __shared__ float s_cb[CB_ELEMS];
  const int tid = threadIdx.x;

  // ---- stage 32 KB of codebooks into LDS: TDM async copy (one wave issues) ----
#if defined(__gfx1250__) && __has_builtin(__builtin_amdgcn_tensor_load_to_lds) && \
    __has_builtin(__builtin_amdgcn_s_wait_tensorcnt)
  if (tid < 32) {
    const unsigned lds_off = (unsigned)(uintptr_t)(&s_cb[0]);       // low 32b = LDS offset
    const unsigned long long ga = (unsigned long long)(uintptr_t)cb;
    u32x4 g0;
    g0[0] = 1u;                                                     // count=1 (valid D#)
    g0[1] = lds_off;                                                // lds_addr [63:32]
    g0[2] = (unsigned)ga;                                           // global_addr [95:64]
    g0[3] = ((unsigned)(ga >> 32) & 0x01FFFFFFu) | (2u << 30);      // addr[56:32] | type=2
    i32x8 g1;
    g1[0] = 0x00020000;                                             // data_size=2 -> 4 B
    g1[1] = (int)((CB_ELEMS & 0xFFFF) << 16);                       // tensor_dim0 lo16 @bit48
    g1[2] = 0x00010000;                                             // tensor_dim1 = 1
    g1[3] = (int)((CB_ELEMS & 0xFFFF) << 16);                       // tile_dim0 @bit112
    g1[4] = 0;                                                      // tile_dim1/2 unused
    g1[5] = CB_ELEMS;                                               // tensor_dim0_stride
    g1[6] = 0;
    g1[7] = 0;
    i32x4 z4 = {0, 0, 0, 0};                                        // groups 2/3 unused (<=2D)
    i32x8 z8 = {0, 0, 0, 0, 0, 0, 0, 0};                            // extra group (6-arg form)
    __builtin_amdgcn_tensor_load_to_lds(g0, g1, z4, z4, z8, 0);
    __builtin_amdgcn_s_wait_tensorcnt(0);
  }
  __syncthreads();
#else
  for (int i = tid; i < CB_ELEMS; i += 256) s_cb[i] = cb[i];
  __syncthreads();
#endif

  const int lane = tid & 31;
  const int ln   = lane & 15;
  const bool hi  = lane >= 16;
  const int base = blockIdx.x * 256 + (tid >> 5) * 32;   // 32 tokens per wave
  const int tokMine = base + lane;

  // clamped token indices so EXEC stays all-ones (WMMA requirement)
  const int tA = min(base + ln,      ntok - 1);
  const int tB = min(base + 16 + ln, ntok - 1);

  float h0, h1;
#if defined(__gfx1250__) && __has_builtin(__builtin_amdgcn_wmma_f32_16x16x4_f32)
  // A (16x4, M=rows of enc_w0 zero-padded): lanes 0-15 row m=lane holds K=0,1;
  // lanes 16-31 same rows hold K=2,3.
  v2f a;
  {
    const int kb = hi ? 2 : 0;
    const bool valid = ln < 2;
    a[0] = valid ? ew0[ln * 4 + kb]     : 0.0f;
    a[1] = valid ? ew0[ln * 4 + kb + 1] : 0.0f;
  }
  // B (4x16) = X^T: column n = token; lanes 0-15 hold K=0,1; lanes 16-31 K=2,3.
  v2f b0, b1;
  {
    const float* p = x + (size_t)tA * 4 + (hi ? 2 : 0);
    b0[0] = p[0]; b0[1] = p[1];
    const float* p2 = x + (size_t)tB * 4 + (hi ? 2 : 0);
    b1[0] = p2[0]; b1[1] = p2[1];
  }
  v8f cz = {0.f, 0.f, 0.f, 0.f, 0.f, 0.f, 0.f, 0.f};
  v8f cA = __builtin_amdgcn_wmma_f32_16x16x4_f32(false, a, false, b0, (short)0, cz, false, false);
  v8f cB = __builtin_amdgcn_wmma_f32_16x16x4_f32(false, a, false, b1, (short)0, cz, false, false);
  // C layout: VGPR r, lanes 0-15 -> (M=r, N=lane). h(m=0/1, token n) at lane n, c[0]/c[1].
  const float h0B = __shfl(cB[0], ln, 32);
  const float h1B = __shfl(cB[1], ln, 32);
  h0 = hi ? h0B : cA[0];
  h1 = hi ? h1B : cA[1];
#else
  {
    const int t = min(tokMine, ntok - 1);
    const float x0 = x[(size_t)t * 4 + 0], x1 = x[(size_t)t * 4 + 1];
    const float x2 = x[(size_t)t * 4 + 2], x3 = x[(size_t)t * 4 + 3];
    h0 = ew0[0] * x0 + ew0[1] * x1 + ew0[2] * x2 + ew0[3] * x3;
    h1 = ew0[4] * x0 + ew0[5] * x1 + ew0[6] * x2 + ew0[7] * x3;
  }
#endif
  h0 = fmaxf(h0 + eb0[0], 0.0f);
  h1 = fmaxf(h1 + eb0[1], 0.0f);
  const float z = fmaf(ew1[0], h0, fmaf(ew1[1], h1, eb1[0]));

  // ---- residual VQ: 8 quantizers over scalar codebooks in LDS ----
  float r = z, qsum = 0.0f;
#pragma unroll
  for (int q = 0; q < NQ; ++q) {
    const float* s = s_cb + q * NK;
    float c;
    if (SORTED) {
      // count elements <= r (branchless binary search on pow2-sized array)
      int idx = 0;
#pragma unroll
      for (int step = NK >> 1; step > 0; step >>= 1)
        if (s[idx + step - 1] <= r) idx += step;
      if (idx == 0)       c = s[0];
      else if (idx == NK) c = s[NK - 1];
      else {
        const float lo = s[idx - 1], hv = s[idx];
        c = ((r - lo) <= (hv - r)) ? lo : hv;
      }
    } else {
      float bestd = 3.402823466e38f, bestc = s[0];
      for (int k = 0; k < NK; ++k) {
        const float ck = s[k];
        const float d = (r - ck) * (r - ck);
        if (d < bestd) { bestd = d; bestc = ck; }
      }
      c = bestc;
    }
    qsum += c;
    r -= c;
  }

  // ---- decoder: Linear(1,2)+ReLU -> Linear(2,4) ----
  const float u0 = fmaxf(fmaf(dw0[0], qsum, db0[0]), 0.0f);
  const float u1 = fmaxf(fmaf(dw0[1], qsum, db0[1]), 0.0f);
  if (tokMine < ntok) {
    float4 o;
    o.x = fmaf(dw1[0], u0, fmaf(dw1[1], u1, db1[0]));
    o.y = fmaf(dw1[2], u0, fmaf(dw1[3], u1, db1[1]));
    o.z = fmaf(dw1[4], u0, fmaf(dw1[5], u1, db1[2]));
    o.w = fmaf(dw1[6], u0, fmaf(dw1[7], u1, db1[3]));
    *(float4*)(out + (size_t)tokMine * 4) = o;
  }
}

// ------------------------------- launcher ----------------------------------
extern "C" void kernel_launch(void* const* d_in, const int* in_sizes, int n_in,
                              void* d_out, int out_size, void* d_ws, size_t ws_size,
                              hipStream_t stream) {
  const float* x   = (const float*)d_in[0];
  const float* ew0 = (const float*)d_in[1];
  const float* eb0 = (const float*)d_in[2];
  const float* ew1 = (const float*)d_in[3];
  const float* eb1 = (const float*)d_in[4];
  const float* dw0 = (const float*)d_in[5];
  const float* db0 = (const float*)d_in[6];
  const float* dw1 = (const float*)d_in[7];
  const float* db1 = (const float*)d_in[8];
  const float* cb  = (const float*)d_in[9];
  float* out = (float*)d_out;

  const int ntok   = in_sizes[0] / 4;            // B*N = 131072
  const int blocks = (ntok + 255) / 256;

  if (ws_size >= CB_ELEMS * sizeof(float)) {
    float* sorted = (float*)d_ws;
    rq_sort_codebooks<<<NQ, NK, 0, stream>>>(cb, sorted);
    rq_autoenc<true><<<blocks, 256, 0, stream>>>(x, ew0, eb0, ew1, eb1,
                                                 dw0, db0, dw1, db1, sorted, out, ntok);
  } else {
    rq_autoenc<false><<<blocks, 256, 0, stream>>>(x, ew0, eb0, ew1, eb1,
                                                  dw0, db0, dw1, db1, cb, out, ntok);
  }
}